// SOL_3332894622153
// MI455X (gfx1250) — compile-verified
//
#include <hip/hip_runtime.h>
#include <math.h>

typedef float v2f __attribute__((ext_vector_type(2)));
typedef float v8f __attribute__((ext_vector_type(8)));

#define SDIM  256
#define BATCH 8
#define WID   64
#define CINC  3
#define MODES 32
#define P34   34
#define BSPEC_STRIDE (P34*P34)
#define IMG   (SDIM*SDIM)

__device__ __forceinline__ v8f wmma_f32_k4(v8f acc, v2f a, v2f b) {
  // D = A(16x4) * B(4x16) + C, fp32 in/out
  return __builtin_amdgcn_wmma_f32_16x16x4_f32(false, a, false, b, (short)0, acc, false, false);
}

__device__ __forceinline__ float gelu_exact(float v) {
  return 0.5f * v * (1.0f + erff(v * 0.70710678118654752f));
}

__device__ __forceinline__ int imin(int a, int b) { return a < b ? a : b; }

// ---------------------------------------------------------------- DCT matrix
__global__ void k_dct_init(float* dct, float* dctT) {
  int k = blockIdx.x, x = threadIdx.x;
  double s = sqrt(2.0 / (double)SDIM);
  if (k == 0) s *= 0.70710678118654752440;
  double v = s * cos(3.14159265358979323846 * (x + 0.5) * k / (double)SDIM);
  dct[k * SDIM + x]  = (float)v;
  dctT[x * SDIM + k] = (float)v;
}

// ------------------------------------------------- x (B,S,S,C) -> h (B,C,S,S)
__global__ void k_transpose_in(const float* __restrict__ x, float* __restrict__ h) {
  int idx = blockIdx.x * blockDim.x + threadIdx.x;
  if (idx >= BATCH * CINC * IMG) return;
  int p = idx % IMG;
  int c = (idx / IMG) % CINC;
  int b = idx / (CINC * IMG);
  h[(size_t)(b * WID + c) * IMG + p] = x[((size_t)b * IMG + p) * CINC + c];
}

// ---------------------- truncated forward DCT: bspec[34x34] = (D48 U D48^T)[:34,:34]
__global__ void k_spec_fwd(const float* __restrict__ h, const float* __restrict__ dctT,
                           float* __restrict__ bspec, int cin) {
  extern __shared__ float t1[];                 // 48 x 257 (padded stride)
  const int T1S = 257;
  int b = blockIdx.x / cin, c = blockIdx.x % cin;
  const float* img = h + (size_t)(b * WID + c) * IMG;
  int lane = threadIdx.x & 31, wave = threadIdx.x >> 5;
  int half = lane >> 4, q = lane & 15, k0 = 2 * half;

  // stage 1: T1[48x256] = D48 * U   (K = 256)
  for (int t = wave; t < 48; t += 8) {
    int m0 = (t >> 4) * 16;
    int n0 = (t & 15) * 16;
    v8f acc = {};
    for (int kk = 0; kk < SDIM; kk += 4) {
      v2f a, bb;
      a[0]  = dctT[(kk + k0    ) * SDIM + m0 + q];   // D[m][k], lane-coalesced
      a[1]  = dctT[(kk + k0 + 1) * SDIM + m0 + q];
      bb[0] = img[(size_t)(kk + k0    ) * SDIM + n0 + q];
      bb[1] = img[(size_t)(kk + k0 + 1) * SDIM + n0 + q];
      // branchless stream-ahead on the activation rows (index clamped; harmless dup)
      int kpf = imin(kk + 64 + k0, SDIM - 1);
      __builtin_prefetch(&img[(size_t)kpf * SDIM + n0 + q], 0, 1);
      acc = wmma_f32_k4(acc, a, bb);
    }
    for (int r = 0; r < 8; ++r)
      t1[(m0 + r + 8 * half) * T1S + n0 + q] = acc[r];
  }
  __syncthreads();

  // stage 2: B48 = T1 * D48^T ; keep 34x34 corner
  float* out = bspec + (size_t)(b * cin + c) * BSPEC_STRIDE;
  for (int t = wave; t < 9; t += 8) {
    int m0 = (t / 3) * 16;
    int n0 = (t % 3) * 16;
    v8f acc = {};
    for (int kk = 0; kk < SDIM; kk += 4) {
      v2f a, bb;
      a[0]  = t1[(m0 + q) * T1S + kk + k0];
      a[1]  = t1[(m0 + q) * T1S + kk + k0 + 1];
      bb[0] = dctT[(kk + k0    ) * SDIM + n0 + q];   // D^T[k][n] = D[n][k]
      bb[1] = dctT[(kk + k0 + 1) * SDIM + n0 + q];
      acc = wmma_f32_k4(acc, a, bb);
    }
    int l = n0 + q;
    if (l < P34)
      for (int r = 0; r < 8; ++r) {
        int m = m0 + r + 8 * half;
        if (m < P34) out[m * P34 + l] = acc[r];
      }
  }
}

// --------- per-spectral-bin GEMM: om[b,o,x] = sum_i xp[b,i,x] * w[i,o,x]
// batch (8) padded to M=16; weights are the streaming bottleneck (604MB total),
// so run a branchless global_prefetch stream ~64 K-steps ahead of consumption.
__global__ void k_patch_gemm(const float* __restrict__ bspec, const float* __restrict__ w,
                             float* __restrict__ om, int cin, int cout) {
  int x = blockIdx.x;                       // spectral bin 0..1023
  int r0 = x >> 5, s0 = x & 31;
  int lane = threadIdx.x & 31, wave = threadIdx.x >> 5;
  int half = lane >> 4, q = lane & 15, k0 = 2 * half;
  int n0 = wave * 16;                       // output-channel tile
  int K = cin * 9;
  int opf = imin(n0 + q, cout - 1);         // clamped weight column for prefetch
  v8f acc = {};
  for (int kk = 0; kk < K; kk += 4) {
    v2f a, bb;
#pragma unroll
    for (int j = 0; j < 2; ++j) {
      int i = kk + k0 + j;
      float av = 0.f, bv = 0.f;
      if (i < K) {
        if (q < 8) {                        // valid batch rows only
          int c = i / 9, rem = i % 9;
          av = bspec[(size_t)(q * cin + c) * BSPEC_STRIDE + (r0 + rem / 3) * P34 + (s0 + rem % 3)];
        }
        int o = n0 + q;
        if (o < cout) bv = w[((size_t)i * cout + o) * 1024 + x];
      }
      a[j] = av; bb[j] = bv;
    }
    // branchless speculative prefetch, distance 64 i-values (~16 iterations)
    int ipf = imin(kk + 64 + k0, K - 1);
    __builtin_prefetch(&w[((size_t)ipf * cout + opf) * 1024 + x], 0, 1);
    acc = wmma_f32_k4(acc, a, bb);
  }
  int o = n0 + q;
  if (half == 0 && o < cout)
    for (int r = 0; r < 8; ++r)             // rows 0..7 = batch
      om[((size_t)r * cout + o) * 1024 + x] = acc[r];
}

// ---------------------------------------------- conv1x1: t = pw * h + pb
__global__ void k_conv1x1(const float* __restrict__ h, const float* __restrict__ pw,
                          const float* __restrict__ pb, float* __restrict__ t) {
  int b = blockIdx.z;
  int m0 = blockIdx.y * 16;
  int lane = threadIdx.x & 31, wave = threadIdx.x >> 5;
  int half = lane >> 4, q = lane & 15, k0 = 2 * half;
  int n0 = blockIdx.x * 128 + wave * 16;
  const float* hb = h + (size_t)b * WID * IMG;
  v8f acc = {};
  for (int kk = 0; kk < WID; kk += 4) {
    v2f a, bb;
    a[0]  = pw[(m0 + q) * WID + kk + k0];
    a[1]  = pw[(m0 + q) * WID + kk + k0 + 1];
    bb[0] = hb[(size_t)(kk + k0    ) * IMG + n0 + q];
    bb[1] = hb[(size_t)(kk + k0 + 1) * IMG + n0 + q];
    int kpf = imin(kk + 16 + k0, WID - 1);  // branchless stream-ahead
    __builtin_prefetch(&hb[(size_t)kpf * IMG + n0 + q], 0, 1);
    acc = wmma_f32_k4(acc, a, bb);
  }
  for (int r = 0; r < 8; ++r) {
    int o = m0 + r + 8 * half;
    t[((size_t)(b * WID + o)) * IMG + n0 + q] = acc[r] + pb[o];
  }
}

// ---- inverse truncated DCT + fused epilogue:
//   mode 0: h[b,3+o] = gelu(spec)           (layer 0 concat path)
//   mode 1: h[b,o]  += gelu(tconv + spec)   (residual layers)
__global__ void k_spec_inv(const float* __restrict__ om, const float* __restrict__ dct,
                           const float* __restrict__ tconv, float* __restrict__ h,
                           int cout, int mode) {
  extern __shared__ float P[];                  // 256 x 33 (padded stride)
  const int PS = 33;
  int b = blockIdx.x / cout, o = blockIdx.x % cout;
  const float* omb = om + (size_t)(b * cout + o) * 1024;
  int lane = threadIdx.x & 31, wave = threadIdx.x >> 5;
  int half = lane >> 4, q = lane & 15, k0 = 2 * half;

  // stage 1: P[256x32] = D^T * om32   (K = 32)
  for (int t = wave; t < 32; t += 8) {
    int m0 = (t >> 1) * 16;
    int n0 = (t & 1) * 16;
    v8f acc = {};
    for (int kk = 0; kk < MODES; kk += 4) {
      v2f a, bb;
      a[0]  = dct[(kk + k0    ) * SDIM + m0 + q];   // D^T[m][k] = D[k][m]
      a[1]  = dct[(kk + k0 + 1) * SDIM + m0 + q];
      bb[0] = omb[(kk + k0    ) * MODES + n0 + q];
      bb[1] = omb[(kk + k0 + 1) * MODES + n0 + q];
      acc = wmma_f32_k4(acc, a, bb);
    }
    for (int r = 0; r < 8; ++r)
      P[(m0 + r + 8 * half) * PS + n0 + q] = acc[r];
  }
  __syncthreads();

  // stage 2: spec[256x256] = P * D32, fused activation/residual
  int chan = (mode == 0) ? (CINC + o) : o;
  float* hout = h + (size_t)(b * WID + chan) * IMG;
  const float* tc = (mode == 0) ? nullptr : tconv + (size_t)(b * WID + chan) * IMG;
  for (int t = wave; t < 256; t += 8) {
    int m0 = (t >> 4) * 16;
    int n0 = (t & 15) * 16;
    v8f acc = {};
    for (int kk = 0; kk < MODES; kk += 4) {
      v2f a, bb;
      a[0]  = P[(m0 + q) * PS + kk + k0];
      a[1]  = P[(m0 + q) * PS + kk + k0 + 1];
      bb[0] = dct[(kk + k0    ) * SDIM + n0 + q];   // D[l][w]
      bb[1] = dct[(kk + k0 + 1) * SDIM + n0 + q];
      acc = wmma_f32_k4(acc, a, bb);
    }
    for (int r = 0; r < 8; ++r) {
      size_t p = (size_t)(m0 + r + 8 * half) * SDIM + n0 + q;
      if (mode == 0) hout[p] = gelu_exact(acc[r]);
      else           hout[p] = hout[p] + gelu_exact(tc[p] + acc[r]);
    }
  }
}

// ------------------------- final MLP: y = gelu(h^T fc1 + b1) fc2 + b2  (COUT=1)
__global__ void k_mlp(const float* __restrict__ h, const float* __restrict__ fc1w,
                      const float* __restrict__ fc1b, const float* __restrict__ fc2w,
                      const float* __restrict__ fc2b, float* __restrict__ y) {
  __shared__ float zs[128 * 16];
  __shared__ float red[16 * 16];
  int b = blockIdx.y;
  int n0 = blockIdx.x * 16;                 // 16 pixels per block
  int lane = threadIdx.x & 31, wave = threadIdx.x >> 5;
  int half = lane >> 4, q = lane & 15, k0 = 2 * half;
  const float* hb = h + (size_t)b * WID * IMG;
  int m0 = wave * 16;                       // 8 waves cover j = 0..127
  v8f acc = {};
  for (int kk = 0; kk < WID; kk += 4) {
    v2f a, bb;
    a[0]  = fc1w[(kk + k0    ) * 128 + m0 + q];     // A[j][i] = fc1w[i][j]
    a[1]  = fc1w[(kk + k0 + 1) * 128 + m0 + q];
    bb[0] = hb[(size_t)(kk + k0    ) * IMG + n0 + q];
    bb[1] = hb[(size_t)(kk + k0 + 1) * IMG + n0 + q];
    acc = wmma_f32_k4(acc, a, bb);
  }
  for (int r = 0; r < 8; ++r) {
    int j = m0 + r + 8 * half;
    zs[j * 16 + q] = gelu_exact(acc[r] + fc1b[j]);
  }
  __syncthreads();
  int pcol = threadIdx.x & 15;
  int grp  = threadIdx.x >> 4;              // 16 groups of 8 j-values
  float part = 0.f;
  for (int g = 0; g < 8; ++g) {
    int j = grp * 8 + g;
    part += zs[j * 16 + pcol] * fc2w[j];
  }
  red[grp * 16 + pcol] = part;
  __syncthreads();
  if (threadIdx.x < 16) {
    float s = fc2b[0];
    for (int g2 = 0; g2 < 16; ++g2) s += red[g2 * 16 + threadIdx.x];
    y[(size_t)b * IMG + n0 + threadIdx.x] = s;
  }
}

// ------------- Y = op(D) * X  (right==0)  or  Y = X * op(D)  (right==1)
__global__ void k_dmul(const float* __restrict__ dct, const float* __restrict__ X,
                       float* __restrict__ Y, int right, int trans) {
  size_t base = (size_t)blockIdx.z * IMG;
  int m0 = blockIdx.y * 16, n0 = blockIdx.x * 16;
  int lane = threadIdx.x & 31;
  int half = lane >> 4, q = lane & 15, k0 = 2 * half;
  v8f acc = {};
  for (int kk = 0; kk < SDIM; kk += 4) {
    v2f a, bb;
#pragma unroll
    for (int j = 0; j < 2; ++j) {
      int k = kk + k0 + j;
      float av, bv;
      if (!right) {
        av = trans ? dct[k * SDIM + m0 + q] : dct[(m0 + q) * SDIM + k];
        bv = X[base + (size_t)k * SDIM + n0 + q];
      } else {
        av = X[base + (size_t)(m0 + q) * SDIM + k];
        bv = trans ? dct[(n0 + q) * SDIM + k] : dct[k * SDIM + n0 + q];
      }
      a[j] = av; bb[j] = bv;
    }
    acc = wmma_f32_k4(acc, a, bb);
  }
  for (int r = 0; r < 8; ++r)
    Y[base + (size_t)(m0 + r + 8 * half) * SDIM + n0 + q] = acc[r];
}

extern "C" void kernel_launch(void* const* d_in, const int* in_sizes, int n_in,
                              void* d_out, int out_size, void* d_ws, size_t ws_size,
                              hipStream_t stream) {
  const float* x    = (const float*)d_in[0];
  const float* wl   = (const float*)d_in[1];
  const float* wc[4] = {(const float*)d_in[2], (const float*)d_in[3],
                        (const float*)d_in[4], (const float*)d_in[5]};
  const float* pw[4] = {(const float*)d_in[6], (const float*)d_in[8],
                        (const float*)d_in[10], (const float*)d_in[12]};
  const float* pb[4] = {(const float*)d_in[7], (const float*)d_in[9],
                        (const float*)d_in[11], (const float*)d_in[13]};
  const float* fc1w = (const float*)d_in[14];
  const float* fc1b = (const float*)d_in[15];
  const float* fc2w = (const float*)d_in[16];
  const float* fc2b = (const float*)d_in[17];
  float* out = (float*)d_out;

  float* ws    = (float*)d_ws;
  float* dct   = ws;                                   // 65536
  float* dctT  = dct + 65536;                          // 65536
  float* h     = dctT + 65536;                         // 8*64*65536
  float* t     = h + (size_t)BATCH * WID * IMG;        // 8*64*65536
  float* bspec = t + (size_t)BATCH * WID * IMG;        // 8*64*1156
  float* om    = bspec + (size_t)BATCH * WID * BSPEC_STRIDE;  // 8*64*1024
  float* y0    = om + (size_t)BATCH * WID * 1024;      // 8*65536
  float* y1    = y0 + (size_t)BATCH * IMG;             // 8*65536

  const size_t ldsF = 48 * 257 * sizeof(float);        // spec_fwd T1
  const size_t ldsI = 256 * 33 * sizeof(float);        // spec_inv P

  k_dct_init<<<dim3(256), dim3(256), 0, stream>>>(dct, dctT);
  k_transpose_in<<<dim3((BATCH * CINC * IMG + 255) / 256), dim3(256), 0, stream>>>(x, h);

  // layer 0: h = concat(x_t, gelu(pseudo_spectra(x_t, wl, 61)))
  k_spec_fwd<<<dim3(BATCH * CINC), dim3(256), ldsF, stream>>>(h, dctT, bspec, CINC);
  k_patch_gemm<<<dim3(1024), dim3(128), 0, stream>>>(bspec, wl, om, CINC, WID - CINC);
  k_spec_inv<<<dim3(BATCH * (WID - CINC)), dim3(256), ldsI, stream>>>(om, dct, nullptr, h, WID - CINC, 0);

  // 4 residual spectral layers
  for (int l = 0; l < 4; ++l) {
    k_spec_fwd<<<dim3(BATCH * WID), dim3(256), ldsF, stream>>>(h, dctT, bspec, WID);
    k_patch_gemm<<<dim3(1024), dim3(128), 0, stream>>>(bspec, wc[l], om, WID, WID);
    k_conv1x1<<<dim3(512, 4, BATCH), dim3(256), 0, stream>>>(h, pw[l], pb[l], t);
    k_spec_inv<<<dim3(BATCH * WID), dim3(256), ldsI, stream>>>(om, dct, t, h, WID, 1);
  }

  // per-pixel MLP 64 -> 128 -> 1
  k_mlp<<<dim3(4096, BATCH), dim3(256), 0, stream>>>(h, fc1w, fc1b, fc2w, fc2b, y0);

  // y = T_fwd(T_inv2(y)) : D * (D^T y D) * D^T
  k_dmul<<<dim3(16, 16, BATCH), dim3(32), 0, stream>>>(dct, y0, y1, 0, 1); // D^T * y
  k_dmul<<<dim3(16, 16, BATCH), dim3(32), 0, stream>>>(dct, y1, y0, 1, 0); //      * D
  k_dmul<<<dim3(16, 16, BATCH), dim3(32), 0, stream>>>(dct, y0, y1, 0, 0); // D *
  k_dmul<<<dim3(16, 16, BATCH), dim3(32), 0, stream>>>(dct, y1, out, 1, 1); //     * D^T
}